// SLAYLaplaceAttention_5377299054796
// MI455X (gfx1250) — compile-verified
//
#include <hip/hip_runtime.h>

// ---------------------------------------------------------------- constants
#define B_    4
#define L_    4096
#define E_    1024
#define H_    16
#define D_    64
#define M_    64
#define R_    2
#define F_    128          // R*M
#define TOK   (B_ * L_)    // 16384
#define QKV_N (3 * E_)     // 3072

#define XS_LD  68          // 128 x 64 token tile (float4-stored -> mult of 4)
#define FS_LD  130         // 128 x 128 feature tile (scalar store, v2f read)
#define OMT_LD 66          // omega^T per r: [m][d] (scalar store, v2f read)
#define VT_LD  130         // v^T / kv^T tile: [d][token] (scalar store, v2f read)
#define AS_LD  36          // GEMM A tile stride (float4-stored -> mult of 4)
#define WT_LD  38          // GEMM W^T tile stride (scalar store, v2f read)

typedef float v2f __attribute__((ext_vector_type(2)));
typedef float v8f __attribute__((ext_vector_type(8)));

// =======================================================================
// Generic fp32 WMMA GEMM:  C[Mtot,N] = A[Mtot,K] @ W[K,N] + bias
// block = 256 threads (8 waves); block tile 128(M) x 64(N); K-step 32.
// A staged row-major (K-minor), W staged transposed (K-minor) so both
// WMMA fragments are single aligned ds_load_b64s.
// =======================================================================
__global__ __launch_bounds__(256)
void gemm_f32_wmma(const float* __restrict__ A, const float* __restrict__ W,
                   const float* __restrict__ bias, float* __restrict__ Cout,
                   int Mtot, int K, int N) {
  __shared__ float As[128 * AS_LD];
  __shared__ float WT[64 * WT_LD];
  const int tid  = threadIdx.x;
  const int lane = tid & 31;
  const int wave = tid >> 5;
  const int half = lane >> 4;
  const int lm   = lane & 15;
  const int m0 = blockIdx.x * 128;
  const int n0 = blockIdx.y * 64;

  v8f acc[4] = {};

  for (int k0 = 0; k0 < K; k0 += 32) {
    if (k0 + 32 < K)
      __builtin_prefetch(A + (size_t)(m0 + (tid >> 1)) * K + k0 + 32, 0, 0);
    // stage A tile 128x32 (1024 float4, 4 per thread), row-major
#pragma unroll
    for (int i = 0; i < 4; ++i) {
      int idx = tid + i * 256;
      int r = idx >> 3, c4 = (idx & 7) << 2;
      float4 v = *(const float4*)(A + (size_t)(m0 + r) * K + (k0 + c4));
      *(float4*)(As + r * AS_LD + c4) = v;
    }
    // stage W tile 32x64 transposed -> WT[n][k]
#pragma unroll
    for (int i = 0; i < 2; ++i) {
      int idx = tid + i * 256;
      int r = idx >> 4, c4 = (idx & 15) << 2;   // r = k row, c4 = n col
      float4 v = *(const float4*)(W + (size_t)(k0 + r) * N + (n0 + c4));
      WT[(c4 + 0) * WT_LD + r] = v.x;
      WT[(c4 + 1) * WT_LD + r] = v.y;
      WT[(c4 + 2) * WT_LD + r] = v.z;
      WT[(c4 + 3) * WT_LD + r] = v.w;
    }
    __syncthreads();
    for (int kk = 0; kk < 32; kk += 4) {
      int krow = kk + 2 * half;
      v2f a = *(const v2f*)(As + (wave * 16 + lm) * AS_LD + krow);
#pragma unroll
      for (int nt = 0; nt < 4; ++nt) {
        v2f b = *(const v2f*)(WT + (nt * 16 + lm) * WT_LD + krow);
        acc[nt] = __builtin_amdgcn_wmma_f32_16x16x4_f32(
            false, a, false, b, (short)0, acc[nt], false, false);
      }
    }
    __syncthreads();
  }
  // C/D layout: VGPR i -> row = i + 8*half, col = nt*16 + lm
#pragma unroll
  for (int nt = 0; nt < 4; ++nt)
#pragma unroll
    for (int i = 0; i < 8; ++i) {
      int row = m0 + wave * 16 + i + 8 * half;
      int col = n0 + nt * 16 + lm;
      Cout[(size_t)row * N + col] = acc[nt][i] + bias[col];
    }
}

// =======================================================================
// Feature pipeline: 128 tokens x D=64 head slice -> normalize, project
// through omega^T (WMMA), PRF nonlinearity -> Fs.
// TRANS=false : Fs[token][f]  (token-major, for ctx A operand)
// TRANS=true  : Fs[f][token]  (f-major,     for kv  A operand)
// =======================================================================
template <bool TRANS>
__device__ __forceinline__ void compute_features(
    const float* __restrict__ src,      // 128 rows, row stride QKV_N
    const float* s, const float* sq2s, const float* scale,
    float* __restrict__ Xs, float* __restrict__ Fs,
    const float* __restrict__ OmT, float* __restrict__ ns) {
  const int tid  = threadIdx.x;
  const int lane = tid & 31;
  const int wave = tid >> 5;
  const int half = lane >> 4;
  const int lm   = lane & 15;

  // load 128 x 64 token tile (row-major, K-minor for the A operand)
  for (int idx = tid; idx < 2048; idx += 256) {
    int r = idx >> 4, c4 = (idx & 15) << 2;
    float4 v = *(const float4*)(src + (size_t)r * QKV_N + c4);
    *(float4*)(Xs + r * XS_LD + c4) = v;
  }
  __syncthreads();
  // per-row L2 norm, inv with EPS clip
  if (tid < 128) {
    float ss = 0.f;
#pragma unroll
    for (int c = 0; c < 64; ++c) { float x = Xs[tid * XS_LD + c]; ss += x * x; }
    ns[tid] = 1.0f / fmaxf(sqrtf(ss), 1e-6f);
  }
  __syncthreads();
  for (int idx = tid; idx < 8192; idx += 256) {
    int r = idx >> 6, c = idx & 63;
    Xs[r * XS_LD + c] *= ns[r];
  }
  __syncthreads();
  // proj[r] = x_norm @ omega[r] : both fragments contiguous ds_load_b64
  v8f pa[2][4] = {};
  for (int kk = 0; kk < 64; kk += 4) {
    int krow = kk + 2 * half;
    v2f a = *(const v2f*)(Xs + (wave * 16 + lm) * XS_LD + krow);
#pragma unroll
    for (int r = 0; r < 2; ++r)
#pragma unroll
      for (int nt = 0; nt < 4; ++nt) {
        v2f b = *(const v2f*)(OmT + (r * 64 + nt * 16 + lm) * OMT_LD + krow);
        pa[r][nt] = __builtin_amdgcn_wmma_f32_16x16x4_f32(
            false, a, false, b, (short)0, pa[r][nt], false, false);
      }
  }
  __syncthreads();  // all Xs reads done (caller may reuse Xs)
  // PRF nonlinearity
#pragma unroll
  for (int r = 0; r < 2; ++r)
#pragma unroll
    for (int nt = 0; nt < 4; ++nt)
#pragma unroll
      for (int i = 0; i < 8; ++i) {
        int row = wave * 16 + i + 8 * half;
        int f   = r * 64 + nt * 16 + lm;
        float arg = fminf(fmaxf(pa[r][nt][i] * sq2s[r] - s[r], -10.f), 10.f);
        float val = __expf(arg) * scale[r];
        if (TRANS) Fs[f * FS_LD + row] = val;
        else       Fs[row * FS_LD + f] = val;
      }
  __syncthreads();
}

// stage omega transposed: OmT[r][m][d]  (global reads coalesced over m)
__device__ __forceinline__ void stage_omega_t(const float* __restrict__ omega,
                                              int h, float* __restrict__ OmT) {
  for (int i = threadIdx.x; i < 8192; i += 256) {
    int r = i >> 12, dk = (i >> 6) & 63, m = i & 63;
    OmT[(r * 64 + m) * OMT_LD + dk] =
        omega[(((size_t)r * H_ + h) * 64 + dk) * 64 + m];
  }
}

// =======================================================================
// kv_kernel: per (b,h, 128-token chunk) accumulate kv[F,D] and ksum[F]
// =======================================================================
__global__ __launch_bounds__(256)
void kv_kernel(const float* __restrict__ qkv, const float* __restrict__ omega,
               const float* __restrict__ qn, const float* __restrict__ qw,
               float* __restrict__ kv, float* __restrict__ ksum) {
  extern __shared__ float smem[];
  float* buf0 = smem;                     // Xs (128*XS_LD) then Vt (64*VT_LD)
  float* Fs   = buf0 + 128 * XS_LD;       // FsT: [f][token]
  float* OmT  = Fs + 128 * FS_LD;
  float* ns   = OmT + 2 * 64 * OMT_LD;

  const int tid  = threadIdx.x;
  const int lane = tid & 31, wave = tid >> 5;
  const int half = lane >> 4, lm = lane & 15;
  const int bh = blockIdx.x, b = bh >> 4, h = bh & 15;
  const int l0 = blockIdx.y * 128;

  stage_omega_t(omega, h, OmT);
  float s[2], sq[2], sc[2];
#pragma unroll
  for (int r = 0; r < 2; ++r) {
    s[r]  = qn[r];
    sq[r] = sqrtf(2.f * fmaxf(s[r], 0.f));
    sc[r] = sqrtf(fmaxf(qw[r], 0.f)) * 0.125f;   // / sqrt(M=64)
  }
  __syncthreads();

  const float* kbase = qkv + ((size_t)(b * L_ + l0)) * QKV_N + E_ + h * 64;
  compute_features<true>(kbase, s, sq, sc, buf0, Fs, OmT, ns);

  // reuse buf0 as Vt[d][token] (transposed v tile -> contiguous B frags)
  float* Vt = buf0;
  const float* vbase = qkv + ((size_t)(b * L_ + l0)) * QKV_N + 2 * E_ + h * 64;
  for (int idx = tid; idx < 2048; idx += 256) {
    int r = idx >> 4, c4 = (idx & 15) << 2;     // r = token, c4 = d
    float4 v = *(const float4*)(vbase + (size_t)r * QKV_N + c4);
    Vt[(c4 + 0) * VT_LD + r] = v.x;
    Vt[(c4 + 1) * VT_LD + r] = v.y;
    Vt[(c4 + 2) * VT_LD + r] = v.z;
    Vt[(c4 + 3) * VT_LD + r] = v.w;
  }
  // ksum: FsT rows are contiguous in token
  if (tid < 128) {
    float acc = 0.f;
#pragma unroll 8
    for (int r = 0; r < 128; ++r) acc += Fs[tid * FS_LD + r];
    atomicAdd(&ksum[(size_t)bh * F_ + tid], acc);
  }
  __syncthreads();

  // kv += k_feat^T @ v   (wave w owns feature rows [w*16, w*16+16))
  v8f acc[4] = {};
  for (int kb = 0; kb < 128; kb += 4) {
    int krow = kb + 2 * half;
    v2f a = *(const v2f*)(Fs + (wave * 16 + lm) * FS_LD + krow); // FsT[f][tok]
#pragma unroll
    for (int nt = 0; nt < 4; ++nt) {
      v2f bb = *(const v2f*)(Vt + (nt * 16 + lm) * VT_LD + krow);
      acc[nt] = __builtin_amdgcn_wmma_f32_16x16x4_f32(
          false, a, false, bb, (short)0, acc[nt], false, false);
    }
  }
  float* kvb = kv + (size_t)bh * F_ * D_;
#pragma unroll
  for (int nt = 0; nt < 4; ++nt)
#pragma unroll
    for (int i = 0; i < 8; ++i) {
      int fr = wave * 16 + i + 8 * half;
      int d  = nt * 16 + lm;
      atomicAdd(&kvb[fr * 64 + d], acc[nt][i]);
    }
}

// =======================================================================
// ctx_kernel: context = q_feat @ kv, norm = q_feat . ksum, write [B,L,E]
// =======================================================================
__global__ __launch_bounds__(256)
void ctx_kernel(const float* __restrict__ qkv, const float* __restrict__ omega,
                const float* __restrict__ qn, const float* __restrict__ qw,
                const float* __restrict__ kv, const float* __restrict__ ksum,
                float* __restrict__ attn) {
  extern __shared__ float smem[];
  float* buf0 = smem;                     // Xs then KVt[d][f]
  float* Fs   = buf0 + 128 * XS_LD;       // Fs: [token][f]
  float* OmT  = Fs + 128 * FS_LD;
  float* ns   = OmT + 2 * 64 * OMT_LD;
  float* ks   = ns + 128;
  float* nrm  = ks + 128;

  const int tid  = threadIdx.x;
  const int lane = tid & 31, wave = tid >> 5;
  const int half = lane >> 4, lm = lane & 15;
  const int bh = blockIdx.x, b = bh >> 4, h = bh & 15;
  const int l0 = blockIdx.y * 128;

  stage_omega_t(omega, h, OmT);
  float s[2], sq[2], sc[2];
#pragma unroll
  for (int r = 0; r < 2; ++r) {
    s[r]  = qn[r];
    sq[r] = sqrtf(2.f * fmaxf(s[r], 0.f));
    sc[r] = sqrtf(fmaxf(qw[r], 0.f)) * 0.125f;
  }
  __syncthreads();

  const float* qbase = qkv + ((size_t)(b * L_ + l0)) * QKV_N + h * 64;
  compute_features<false>(qbase, s, sq, sc, buf0, Fs, OmT, ns);

  // reuse buf0 as KVt[d][f] (transposed kv tile); load ksum
  float* KVt = buf0;
  const float* kvb = kv + (size_t)bh * F_ * D_;
  for (int idx = tid; idx < 2048; idx += 256) {
    int f = idx >> 4, c4 = (idx & 15) << 2;     // f row, c4 = d
    float4 v = *(const float4*)(kvb + f * 64 + c4);
    KVt[(c4 + 0) * VT_LD + f] = v.x;
    KVt[(c4 + 1) * VT_LD + f] = v.y;
    KVt[(c4 + 2) * VT_LD + f] = v.z;
    KVt[(c4 + 3) * VT_LD + f] = v.w;
  }
  if (tid < 128) ks[tid] = ksum[(size_t)bh * F_ + tid];
  __syncthreads();

  // per-row normalizer  1 / max(q_feat . ksum, 1e-4)
  if (tid < 128) {
    float acc = 0.f;
#pragma unroll 8
    for (int f = 0; f < 128; ++f) acc += Fs[tid * FS_LD + f] * ks[f];
    nrm[tid] = 1.0f / fmaxf(acc, 1e-4f);
  }

  // context = q_feat @ kv  (wave w owns token rows [w*16, w*16+16))
  v8f acc[4] = {};
  for (int kb = 0; kb < 128; kb += 4) {
    int krow = kb + 2 * half;
    v2f a = *(const v2f*)(Fs + (wave * 16 + lm) * FS_LD + krow); // [tok][f]
#pragma unroll
    for (int nt = 0; nt < 4; ++nt) {
      v2f bb = *(const v2f*)(KVt + (nt * 16 + lm) * VT_LD + krow);
      acc[nt] = __builtin_amdgcn_wmma_f32_16x16x4_f32(
          false, a, false, bb, (short)0, acc[nt], false, false);
    }
  }
  __syncthreads();   // nrm visible

  float* ob = attn + ((size_t)(b * L_ + l0)) * E_ + h * 64;
#pragma unroll
  for (int nt = 0; nt < 4; ++nt)
#pragma unroll
    for (int i = 0; i < 8; ++i) {
      int row = wave * 16 + i + 8 * half;
      int d   = nt * 16 + lm;
      ob[(size_t)row * E_ + d] = acc[nt][i] * nrm[row];
    }
}

__global__ void zero_kernel(float* __restrict__ p, int n) {
  int i = blockIdx.x * blockDim.x + threadIdx.x;
  if (i < n) p[i] = 0.f;
}

// =======================================================================
extern "C" void kernel_launch(void* const* d_in, const int* in_sizes, int n_in,
                              void* d_out, int out_size, void* d_ws, size_t ws_size,
                              hipStream_t stream) {
  const float* x      = (const float*)d_in[0];
  const float* qkv_w  = (const float*)d_in[1];
  const float* qkv_b  = (const float*)d_in[2];
  const float* out_w  = (const float*)d_in[3];
  const float* out_b  = (const float*)d_in[4];
  const float* omega  = (const float*)d_in[5];
  const float* qn     = (const float*)d_in[6];
  const float* qw     = (const float*)d_in[7];

  float* ws   = (float*)d_ws;
  float* qkv  = ws;                                  // 16384*3072
  float* kv   = qkv  + (size_t)TOK * QKV_N;          // 64*128*64
  float* ksum = kv   + (size_t)B_ * H_ * F_ * D_;    // 64*128
  float* attn = ksum + (size_t)B_ * H_ * F_;         // 16384*1024

  const size_t fused_lds =
      (128 * XS_LD + 128 * FS_LD + 2 * 64 * OMT_LD + 3 * 128) * sizeof(float);

  {
    int n = B_ * H_ * F_ * D_ + B_ * H_ * F_;
    zero_kernel<<<(n + 255) / 256, 256, 0, stream>>>(kv, n);
  }
  gemm_f32_wmma<<<dim3(TOK / 128, QKV_N / 64), 256, 0, stream>>>(
      x, qkv_w, qkv_b, qkv, TOK, E_, QKV_N);
  kv_kernel<<<dim3(B_ * H_, L_ / 128), 256, fused_lds, stream>>>(
      qkv, omega, qn, qw, kv, ksum);
  ctx_kernel<<<dim3(B_ * H_, L_ / 128), 256, fused_lds, stream>>>(
      qkv, omega, qn, qw, kv, ksum, attn);
  gemm_f32_wmma<<<dim3(TOK / 128, E_ / 64), 256, 0, stream>>>(
      attn, out_w, out_b, (float*)d_out, TOK, E_, E_);
}